// FNO_3D_25795573580623
// MI455X (gfx1250) — compile-verified
//
#include <hip/hip_runtime.h>
#include <hip/hip_bf16.h>

// ---------------------------------------------------------------------------
// FNO-3D on gfx1250 (MI455X).  All DFT stages are truncated-mode dense
// matmuls executed with V_WMMA_F32_16X16X4_F32 (f32 matrix core, wave32).
// Problem is bandwidth-bound (~4.8 GB traffic @ 23.3 TB/s), so full f32
// WMMA costs nothing vs bf16 and keeps reference accuracy.
// The inverse-k2 DFT and the irfft-k3 (+ local channel matmul + GELU) are
// fused through LDS to kill a 168 MB/layer HBM round trip.
// ---------------------------------------------------------------------------

typedef float v2f __attribute__((ext_vector_type(2)));
typedef float v8f __attribute__((ext_vector_type(8)));

#define S1N 256
#define S2N 256
#define S3N 128
#define DV  8
#define K1M 30
#define K2M 30
#define K3M 20

// ---- workspace layout (floats) --------------------------------------------
static constexpr size_t NV = (size_t)S1N * S2N * S3N * DV;        // 67,108,864
static constexpr size_t NA = (size_t)S1N * S2N * K3M * DV;        // 10,485,760
static constexpr size_t NB = (size_t)S1N * 32 * K3M * DV;         // 1,310,720 (k2 padded to 32)
static constexpr size_t NC = (size_t)32 * (K2M * K3M * DV);       // 153,600   (k1 padded to 32)

static constexpr size_t OFF_V   = 0;
static constexpr size_t OFF_AR  = OFF_V  + NV;
static constexpr size_t OFF_AI  = OFF_AR + NA;
static constexpr size_t OFF_BR  = OFF_AI + NA;
static constexpr size_t OFF_BI  = OFF_BR + NB;
static constexpr size_t OFF_CR  = OFF_BI + NB;
static constexpr size_t OFF_CI  = OFF_CR + NC;
static constexpr size_t OFF_DR  = OFF_CI + NC;
static constexpr size_t OFF_DI  = OFF_DR + NC;
static constexpr size_t OFF_ER  = OFF_DI + NC;
static constexpr size_t OFF_EI  = OFF_ER + NB;
static constexpr size_t OFF_F2  = OFF_EI + NB;          // [48][128]  cos | -sin rows
static constexpr size_t OFF_F1C = OFF_F2  + 48 * 128;   // [32][256]
static constexpr size_t OFF_F1S = OFF_F1C + 32 * 256;
static constexpr size_t OFF_F1N = OFF_F1S + 32 * 256;   // -sin
static constexpr size_t OFF_G1C = OFF_F1N + 32 * 256;   // [256][32]  (includes 1/256)
static constexpr size_t OFF_G1S = OFF_G1C + 256 * 32;
static constexpr size_t OFF_G1N = OFF_G1S + 256 * 32;   // -sin/256
static constexpr size_t OFF_G2C = OFF_G1N + 256 * 32;   // [128][20]  (irfft real matrix)
static constexpr size_t OFF_G2S = OFF_G2C + 128 * 20;
static constexpr size_t ZERO_N  = OFF_F2 - OFF_BR;      // pad-carrying spectral buffers

// ---------------------------------------------------------------------------
__device__ __forceinline__ float gelu_exact(float x) {
    return 0.5f * x * (1.0f + erff(x * 0.70710678118654752440f));
}

__device__ __forceinline__ v8f wmma4(v2f a, v2f b, v8f c) {
    // D = A(16x4) * B(4x16) + C   -> v_wmma_f32_16x16x4_f32
    return __builtin_amdgcn_wmma_f32_16x16x4_f32(false, a, false, b, (short)0, c, false, false);
}

// A-fragment of a row-major [rows][ld] matrix, tile origin (row0, k0).
// ISA layout: lanes 0-15 hold K=k0,k0+1; lanes 16-31 hold K=k0+2,k0+3; M = lane&15.
__device__ __forceinline__ v2f frag_a(const float* __restrict__ M, int ld, int row0, int k0, int lane) {
    int m  = row0 + (lane & 15);
    int kk = k0 + ((lane >> 4) << 1);
    v2f a;
    a.x = M[(size_t)m * ld + kk];
    a.y = M[(size_t)m * ld + kk + 1];
    return a;
}

// ---------------------------------------------------------------------------
__global__ void k_init_mats(float* __restrict__ ws) {
    const double TWO_PI = 6.283185307179586476925286766559;
    int t = threadIdx.x;
    float* F2  = ws + OFF_F2;
    float* F1c = ws + OFF_F1C; float* F1s = ws + OFF_F1S; float* F1n = ws + OFF_F1N;
    float* G1c = ws + OFF_G1C; float* G1s = ws + OFF_G1S; float* G1n = ws + OFF_G1N;
    float* G2c = ws + OFF_G2C; float* G2s = ws + OFF_G2S;

    for (int i = t; i < 48 * 128; i += 256) {            // forward rfft-128, 20 modes
        int k = i / 128, n = i % 128;
        double v = 0.0;
        if (k < 20)      v =  cos(TWO_PI * ((k * n) % 128) / 128.0);
        else if (k < 40) v = -sin(TWO_PI * (((k - 20) * n) % 128) / 128.0);
        F2[i] = (float)v;
    }
    for (int i = t; i < 32 * 256; i += 256) {            // forward DFT-256, 30 modes
        int k = i / 256, n = i % 256;
        double c = 0.0, s = 0.0;
        if (k < 30) { double a = TWO_PI * ((k * n) % 256) / 256.0; c = cos(a); s = sin(a); }
        F1c[i] = (float)c; F1s[i] = (float)s; F1n[i] = -(float)s;
    }
    for (int i = t; i < 256 * 32; i += 256) {            // inverse DFT-256 (zero-padded), /256
        int n = i / 32, k = i % 32;
        double c = 0.0, s = 0.0;
        if (k < 30) { double a = TWO_PI * ((k * n) % 256) / 256.0; c = cos(a) / 256.0; s = sin(a) / 256.0; }
        G1c[i] = (float)c; G1s[i] = (float)s; G1n[i] = -(float)s;
    }
    for (int i = t; i < 128 * 20; i += 256) {            // irfft-128 from 20 modes (real out)
        int n = i / 20, k = i % 20;
        double c, s;
        if (k == 0) { c = 1.0 / 128.0; s = 0.0; }        // imag(DC) ignored, as numpy does
        else { double a = TWO_PI * ((k * n) % 128) / 128.0; c = 2.0 * cos(a) / 128.0; s = -2.0 * sin(a) / 128.0; }
        G2c[i] = (float)c; G2s[i] = (float)s;
    }
}

__global__ void k_zero(float* __restrict__ p, size_t n) {
    size_t t = (size_t)blockIdx.x * blockDim.x + threadIdx.x;
    size_t stride = (size_t)gridDim.x * blockDim.x;
    for (size_t i = t; i < n; i += stride) p[i] = 0.0f;
}

__global__ void k_shallow(const float* __restrict__ x, const float* __restrict__ sw,
                          const float* __restrict__ sb, float* __restrict__ v) {
    size_t t = (size_t)blockIdx.x * blockDim.x + threadIdx.x;   // point index < 8,388,608
    float xv = x[t];
    #pragma unroll
    for (int c = 0; c < DV; ++c)
        v[t * DV + c] = gelu_exact(xv * sw[c] + sb[c]);
}

// ---- forward rfft along s3: v[s1,s2,:,c] -> Ar/Ai[s1,s2,k3,c] -------------
// GEMM per (s1, s2-pair): M=40 (20 cos + 20 (-sin) rows, pad 48), K=128, N=16
__global__ void k_stageA(const float* __restrict__ v, const float* __restrict__ F2,
                         float* __restrict__ Ar, float* __restrict__ Ai) {
    const int s2p = blockIdx.x, s1 = blockIdx.y;
    const int wave = threadIdx.x >> 5, lane = threadIdx.x & 31;
    const int n = lane & 15, s2 = s2p * 2 + (n >> 3), c = n & 7;
    const int koff = (lane >> 4) << 1;
    const float* vb = v + ((size_t)(s1 * S2N + s2) * S3N) * DV + c;   // + s3*8
    v8f acc = {0, 0, 0, 0, 0, 0, 0, 0};
    for (int k0 = 0; k0 < 128; k0 += 4) {
        v2f a = frag_a(F2, 128, wave * 16, k0, lane);
        v2f b; b.x = vb[(size_t)(k0 + koff) * DV]; b.y = vb[(size_t)(k0 + koff + 1) * DV];
        acc = wmma4(a, b, acc);
    }
    const int mh = (lane >> 4) << 3;
    #pragma unroll
    for (int r = 0; r < 8; ++r) {
        int g = wave * 16 + r + mh;
        size_t o = ((size_t)(s1 * S2N + s2) * K3M) * DV + c;
        if (g < 20)       Ar[o + (size_t)g * DV]        = acc[r];
        else if (g < 40)  Ai[o + (size_t)(g - 20) * DV] = acc[r];
    }
}

// ---- forward DFT along s2 (complex): A[s1,:,k3,c] -> B[s1,k2,k3,c] --------
// Br = Fc@Ar + Fs@Ai ; Bi = Fc@Ai + (-Fs)@Ar.  M=k2 (2 tiles), K=256, N=16
__global__ void k_stageB(const float* __restrict__ Ar, const float* __restrict__ Ai,
                         const float* __restrict__ Fc, const float* __restrict__ Fs,
                         const float* __restrict__ Fn,
                         float* __restrict__ Br, float* __restrict__ Bi) {
    const int k3p = blockIdx.x, s1 = blockIdx.y;
    const int wave = threadIdx.x >> 5, lane = threadIdx.x & 31;
    const int n = lane & 15, k3 = k3p * 2 + (n >> 3), c = n & 7;
    const int koff = (lane >> 4) << 1;
    const size_t base = (size_t)s1 * S2N * (K3M * DV) + (size_t)k3 * DV + c;  // + s2*160
    v8f ar = {0,0,0,0,0,0,0,0}, ai = {0,0,0,0,0,0,0,0};
    for (int k0 = 0; k0 < 256; k0 += 4) {
        v2f ac  = frag_a(Fc, 256, wave * 16, k0, lane);
        v2f as  = frag_a(Fs, 256, wave * 16, k0, lane);
        v2f an  = frag_a(Fn, 256, wave * 16, k0, lane);
        v2f br_, bi_;
        br_.x = Ar[base + (size_t)(k0 + koff) * 160]; br_.y = Ar[base + (size_t)(k0 + koff + 1) * 160];
        bi_.x = Ai[base + (size_t)(k0 + koff) * 160]; bi_.y = Ai[base + (size_t)(k0 + koff + 1) * 160];
        ar = wmma4(ac, br_, ar);
        ar = wmma4(as, bi_, ar);
        ai = wmma4(ac, bi_, ai);
        ai = wmma4(an, br_, ai);
    }
    const int mh = (lane >> 4) << 3;
    #pragma unroll
    for (int r = 0; r < 8; ++r) {
        int k2 = wave * 16 + r + mh;
        if (k2 < K2M) {
            size_t o = (size_t)s1 * 5120 + (size_t)k2 * 160 + (size_t)k3 * DV + c;
            Br[o] = ar[r]; Bi[o] = ai[r];
        }
    }
}

// ---- forward DFT along s1 (complex): B[:,col] -> C[k1,col], col=(k2,k3,c) -
__global__ void k_stageC(const float* __restrict__ Br, const float* __restrict__ Bi,
                         const float* __restrict__ Fc, const float* __restrict__ Fs,
                         const float* __restrict__ Fn,
                         float* __restrict__ Cr, float* __restrict__ Ci) {
    const int nt = blockIdx.x;                       // 0..299
    const int wave = threadIdx.x >> 5, lane = threadIdx.x & 31;
    const int n = lane & 15, col = nt * 16 + n;
    const int koff = (lane >> 4) << 1;
    v8f cr = {0,0,0,0,0,0,0,0}, ci = {0,0,0,0,0,0,0,0};
    for (int k0 = 0; k0 < 256; k0 += 4) {
        v2f ac = frag_a(Fc, 256, wave * 16, k0, lane);
        v2f as = frag_a(Fs, 256, wave * 16, k0, lane);
        v2f an = frag_a(Fn, 256, wave * 16, k0, lane);
        v2f br_, bi_;
        br_.x = Br[(size_t)(k0 + koff) * 5120 + col]; br_.y = Br[(size_t)(k0 + koff + 1) * 5120 + col];
        bi_.x = Bi[(size_t)(k0 + koff) * 5120 + col]; bi_.y = Bi[(size_t)(k0 + koff + 1) * 5120 + col];
        cr = wmma4(ac, br_, cr);
        cr = wmma4(as, bi_, cr);
        ci = wmma4(ac, bi_, ci);
        ci = wmma4(an, br_, ci);
    }
    const int mh = (lane >> 4) << 3;
    #pragma unroll
    for (int r = 0; r < 8; ++r) {
        int k1 = wave * 16 + r + mh;
        if (k1 < K1M) { Cr[(size_t)k1 * 4800 + col] = cr[r]; Ci[(size_t)k1 * 4800 + col] = ci[r]; }
    }
}

// ---- per-mode 8x8 complex channel mix: D = C @ (Rr + i Ri) ----------------
__global__ void k_spectral(const float* __restrict__ Cr, const float* __restrict__ Ci,
                           const float* __restrict__ Rr, const float* __restrict__ Ri,
                           float* __restrict__ Dr, float* __restrict__ Di) {
    int t = blockIdx.x * blockDim.x + threadIdx.x;
    if (t >= K1M * K2M * K3M * DV) return;
    int e = t & 7, mode = t >> 3;                    // mode = (k1*30+k2)*20+k3
    int k3 = mode % 20, k12 = mode / 20, k2 = k12 % 30, k1 = k12 / 30;
    size_t cb = (size_t)k1 * 4800 + (size_t)k2 * 160 + (size_t)k3 * DV;
    size_t rb = (size_t)mode * 64;
    float sr = 0.f, si = 0.f;
    #pragma unroll
    for (int d = 0; d < 8; ++d) {
        float fr = Cr[cb + d], fi = Ci[cb + d];
        float rr = Rr[rb + d * 8 + e], ri = Ri[rb + d * 8 + e];
        sr += fr * rr - fi * ri;
        si += fr * ri + fi * rr;
    }
    Dr[cb + e] = sr; Di[cb + e] = si;
}

// ---- inverse along k1 -> s1: D[k1,col] -> E[s1,col] -----------------------
// Er = Gc@Dr + (-Gs)@Di ; Ei = Gs@Dr + Gc@Di.  M=s1 (16 tiles), K=32, N=16
__global__ void k_stageCp(const float* __restrict__ Dr, const float* __restrict__ Di,
                          const float* __restrict__ Gc, const float* __restrict__ Gs,
                          const float* __restrict__ Gn,
                          float* __restrict__ Er, float* __restrict__ Ei) {
    const int nt = blockIdx.x, mt = blockIdx.y;
    const int lane = threadIdx.x & 31;
    const int n = lane & 15, col = nt * 16 + n;
    const int koff = (lane >> 4) << 1;
    v8f er = {0,0,0,0,0,0,0,0}, ei = {0,0,0,0,0,0,0,0};
    for (int k0 = 0; k0 < 32; k0 += 4) {
        v2f gc = frag_a(Gc, 32, mt * 16, k0, lane);
        v2f gs = frag_a(Gs, 32, mt * 16, k0, lane);
        v2f gn = frag_a(Gn, 32, mt * 16, k0, lane);
        v2f dr_, di_;
        dr_.x = Dr[(size_t)(k0 + koff) * 4800 + col]; dr_.y = Dr[(size_t)(k0 + koff + 1) * 4800 + col];
        di_.x = Di[(size_t)(k0 + koff) * 4800 + col]; di_.y = Di[(size_t)(k0 + koff + 1) * 4800 + col];
        er = wmma4(gc, dr_, er);
        er = wmma4(gn, di_, er);
        ei = wmma4(gs, dr_, ei);
        ei = wmma4(gc, di_, ei);
    }
    const int mh = (lane >> 4) << 3;
    #pragma unroll
    for (int r = 0; r < 8; ++r) {
        int s1 = mt * 16 + r + mh;
        size_t o = (size_t)s1 * 5120 + col;
        Er[o] = er[r]; Ei[o] = ei[r];
    }
}

// ---- FUSED: inverse k2->s2 (phase 1, into LDS) + irfft k3->s3 + v@w + GELU
// Block: 512 thr = 16 waves, one (s1, 16-wide s2 slab).  LDS holds the
// [16 s2][20 k3][8 c] complex intermediate (20.5 KB of the WGP's 320 KB),
// killing the 168 MB/layer HBM round trip of a separate B' stage.
__global__ void k_fuseBpAp(const float* __restrict__ Er, const float* __restrict__ Ei,
                           const float* __restrict__ G1c, const float* __restrict__ G1s,
                           const float* __restrict__ G1n,
                           const float* __restrict__ G2c, const float* __restrict__ G2s,
                           float* __restrict__ v, const float* __restrict__ w) {
    __shared__ float LFr[16 * 160];
    __shared__ float LFi[16 * 160];
    const int s2t = blockIdx.x;          // s2 slab: s2 in [16*s2t, 16*s2t+16)
    const int s1  = blockIdx.y;
    const int wave = threadIdx.x >> 5, lane = threadIdx.x & 31;
    const int n = lane & 15, c = n & 7;
    const int koff = (lane >> 4) << 1;
    const int mh = (lane >> 4) << 3;

    // phase 1: Fr = Gc@Er + (-Gs)@Ei ; Fi = Gs@Er + Gc@Ei
    // M = s2 (rows s2t*16..+15, one tile), K = k2 (32, pad rows zeroed), N = (k3,c) 160 -> 10 Ntiles
    if (wave < 10) {
        const int k3 = wave * 2 + (n >> 3);
        const size_t eb = (size_t)s1 * 5120 + (size_t)k3 * DV + c;   // + k2*160
        v8f fr = {0,0,0,0,0,0,0,0}, fi = {0,0,0,0,0,0,0,0};
        for (int k0 = 0; k0 < 32; k0 += 4) {
            v2f gc = frag_a(G1c, 32, s2t * 16, k0, lane);
            v2f gs = frag_a(G1s, 32, s2t * 16, k0, lane);
            v2f gn = frag_a(G1n, 32, s2t * 16, k0, lane);
            v2f er_, ei_;
            er_.x = Er[eb + (size_t)(k0 + koff) * 160]; er_.y = Er[eb + (size_t)(k0 + koff + 1) * 160];
            ei_.x = Ei[eb + (size_t)(k0 + koff) * 160]; ei_.y = Ei[eb + (size_t)(k0 + koff + 1) * 160];
            fr = wmma4(gc, er_, fr);
            fr = wmma4(gn, ei_, fr);
            fi = wmma4(gs, er_, fi);
            fi = wmma4(gc, ei_, fi);
        }
        #pragma unroll
        for (int r = 0; r < 8; ++r) {
            int s2l = r + mh;                        // local s2 (row m)
            LFr[s2l * 160 + k3 * DV + c] = fr[r];
            LFi[s2l * 160 + k3 * DV + c] = fi[r];
        }
    }
    __syncthreads();

    // phase 2: acc = G2c@Fr + G2s@Fi (M=s3: 8 tiles; K=20; N=16=(s2 pair, c))
    // 8 s2-pairs x 8 s3-tiles = 64 wave-tiles over 16 waves.
    float wc[8];
    #pragma unroll
    for (int d = 0; d < 8; ++d) wc[d] = w[d * 8 + c];

    for (int t = wave; t < 64; t += 16) {
        const int p  = t & 7;                        // s2 pair within slab
        const int mt = t >> 3;                       // s3 tile
        const int s2l = p * 2 + (n >> 3);
        const int s2  = s2t * 16 + s2l;
        const int fb  = s2l * 160 + c;               // + k3*8 (LDS)
        v8f acc = {0, 0, 0, 0, 0, 0, 0, 0};
        for (int k0 = 0; k0 < 20; k0 += 4) {
            v2f gc = frag_a(G2c, 20, mt * 16, k0, lane);
            v2f gs = frag_a(G2s, 20, mt * 16, k0, lane);
            v2f fr_, fi_;
            fr_.x = LFr[fb + (k0 + koff) * DV]; fr_.y = LFr[fb + (k0 + koff + 1) * DV];
            fi_.x = LFi[fb + (k0 + koff) * DV]; fi_.y = LFi[fb + (k0 + koff + 1) * DV];
            acc = wmma4(gc, fr_, acc);
            acc = wmma4(gs, fi_, acc);
        }
        // epilogue: + v@w, exact GELU, in-place.  All loads precede stores;
        // each point is owned by exactly one wave-tile -> race-free.
        const size_t vrow = (size_t)(s1 * S2N + s2) * S3N;
        float outv[8];
        #pragma unroll
        for (int r = 0; r < 8; ++r) {
            int s3 = mt * 16 + r + mh;
            const float* vb = v + (vrow + s3) * DV;
            float s = acc[r];
            #pragma unroll
            for (int d = 0; d < 8; ++d) s += vb[d] * wc[d];
            outv[r] = gelu_exact(s);
        }
        #pragma unroll
        for (int r = 0; r < 8; ++r) {
            int s3 = mt * 16 + r + mh;
            v[(vrow + s3) * DV + c] = outv[r];
        }
    }
}

__global__ void k_proj(const float* __restrict__ v, const float* __restrict__ pw,
                       const float* __restrict__ pb, float* __restrict__ out) {
    size_t t = (size_t)blockIdx.x * blockDim.x + threadIdx.x;
    float s = pb[0];
    #pragma unroll
    for (int c = 0; c < DV; ++c) s += v[t * DV + c] * pw[c];
    out[t] = s;
}

// ---------------------------------------------------------------------------
extern "C" void kernel_launch(void* const* d_in, const int* in_sizes, int n_in,
                              void* d_out, int out_size, void* d_ws, size_t ws_size,
                              hipStream_t stream) {
    (void)in_sizes; (void)n_in; (void)out_size; (void)ws_size;
    float* ws = (float*)d_ws;
    const float* x  = (const float*)d_in[0];
    const float* sw = (const float*)d_in[1];
    const float* sb = (const float*)d_in[2];
    const float* pw = (const float*)d_in[15];
    const float* pb = (const float*)d_in[16];

    float* V  = ws + OFF_V;
    float* AR = ws + OFF_AR;  float* AI = ws + OFF_AI;
    float* BR = ws + OFF_BR;  float* BI = ws + OFF_BI;
    float* CR = ws + OFF_CR;  float* CI = ws + OFF_CI;
    float* DR = ws + OFF_DR;  float* DI = ws + OFF_DI;
    float* ER = ws + OFF_ER;  float* EI = ws + OFF_EI;
    float* F2  = ws + OFF_F2;
    float* F1C = ws + OFF_F1C; float* F1S = ws + OFF_F1S; float* F1N = ws + OFF_F1N;
    float* G1C = ws + OFF_G1C; float* G1S = ws + OFF_G1S; float* G1N = ws + OFF_G1N;
    float* G2C = ws + OFF_G2C; float* G2S = ws + OFF_G2S;

    k_init_mats<<<1, 256, 0, stream>>>(ws);
    k_zero<<<2048, 256, 0, stream>>>(ws + OFF_BR, ZERO_N);   // pad rows must read as 0
    k_shallow<<<32768, 256, 0, stream>>>(x, sw, sb, V);

    for (int L = 0; L < 4; ++L) {
        const float* Rr = (const float*)d_in[3 + L * 3];
        const float* Ri = (const float*)d_in[4 + L * 3];
        const float* w  = (const float*)d_in[5 + L * 3];
        k_stageA  <<<dim3(128, 256), 96, 0, stream>>>(V, F2, AR, AI);
        k_stageB  <<<dim3(10, 256),  64, 0, stream>>>(AR, AI, F1C, F1S, F1N, BR, BI);
        k_stageC  <<<300,            64, 0, stream>>>(BR, BI, F1C, F1S, F1N, CR, CI);
        k_spectral<<<(K1M * K2M * K3M * DV + 255) / 256, 256, 0, stream>>>(CR, CI, Rr, Ri, DR, DI);
        k_stageCp <<<dim3(300, 16),  32, 0, stream>>>(DR, DI, G1C, G1S, G1N, ER, EI);
        k_fuseBpAp<<<dim3(16, 256), 512, 0, stream>>>(ER, EI, G1C, G1S, G1N, G2C, G2S, V, w);
    }

    k_proj<<<32768, 256, 0, stream>>>(V, pw, pb, (float*)d_out);
}